// LambdaHopGatedGATv2Conv_61942018342917
// MI455X (gfx1250) — compile-verified
//
#include <hip/hip_runtime.h>
#include <math.h>

// ---------------- problem constants (match reference) ----------------
#define NNODES 50000
#define NEDGES 800000
#define DIN    128     // input channels
#define HC     128     // H*C
#define CH     64      // out channels per head
#define NEGSL  0.2f

typedef __attribute__((ext_vector_type(2))) float v2f;
typedef __attribute__((ext_vector_type(8))) float v8f;

// ---------------- fp32 WMMA projection: x_l = x@Wl^T + bl, x_r = x@Wr^T + br
// One block = one 16-row tile of x; 8 waves = 8 col tiles (HC=128 = 8*16).
// A (16x4 f32): lanes 0-15 row M=lane, {K,K+1}; lanes 16-31 row M=lane-16, {K+2,K+3}
// B (4x16 f32): lanes 0-15 col N=lane, {K,K+1}; lanes 16-31 col N=lane-16, {K+2,K+3}
__global__ __launch_bounds__(256)
void proj_wmma(const float* __restrict__ x,
               const float* __restrict__ wl, const float* __restrict__ bl,
               const float* __restrict__ wr, const float* __restrict__ br,
               float* __restrict__ xl, float* __restrict__ xr)
{
    const int lane    = threadIdx.x & 31;
    const int colTile = threadIdx.x >> 5;     // 0..7
    const int rowTile = blockIdx.x;           // 0..3124 (50000 = 16*3125)
    const int row0 = rowTile * 16;
    const int col0 = colTile * 16;
    const int hi   = lane >> 4;               // 0: K,K+1   1: K+2,K+3
    const int l16  = lane & 15;

    const float* xrow  = x  + (size_t)(row0 + l16) * DIN + hi * 2;
    const float* wlrow = wl + (size_t)(col0 + l16) * DIN + hi * 2;
    const float* wrrow = wr + (size_t)(col0 + l16) * DIN + hi * 2;

    v8f accl = {};
    v8f accr = {};
    for (int k = 0; k < DIN; k += 4) {
        v2f a  = *(const v2f*)(xrow  + k);
        v2f bL = *(const v2f*)(wlrow + k);
        v2f bR = *(const v2f*)(wrrow + k);
        accl = __builtin_amdgcn_wmma_f32_16x16x4_f32(false, a, false, bL,
                                                     (short)0, accl, false, false);
        accr = __builtin_amdgcn_wmma_f32_16x16x4_f32(false, a, false, bR,
                                                     (short)0, accr, false, false);
    }
    // C/D layout: VGPR r, lanes 0-15 -> row M=r, lanes 16-31 -> row M=r+8; col = l16
    const float biasl = bl[col0 + l16];
    const float biasr = br[col0 + l16];
#pragma unroll
    for (int r = 0; r < 8; ++r) {
        const int row = row0 + r + hi * 8;
        xl[(size_t)row * HC + col0 + l16] = accl[r] + biasl;
        xr[(size_t)row * HC + col0 + l16] = accr[r] + biasr;
    }
}

// ---------------- gate: softmax(x@gw^T + gb) * lambda(0.5) -> gateS[N,2]
__global__ __launch_bounds__(256)
void gate_kernel(const float* __restrict__ x, const float* __restrict__ gw,
                 const float* __restrict__ gb, float* __restrict__ gateS, int n)
{
    int i = blockIdx.x * blockDim.x + threadIdx.x;
    if (i >= n) return;
    const float4* xr = (const float4*)(x + (size_t)i * DIN);
    const float4* g0 = (const float4*)gw;
    const float4* g1 = (const float4*)(gw + DIN);
    float l0 = 0.f, l1 = 0.f;
#pragma unroll 8
    for (int k = 0; k < DIN / 4; ++k) {
        float4 v = xr[k], a = g0[k], b = g1[k];
        l0 += v.x * a.x + v.y * a.y + v.z * a.z + v.w * a.w;
        l1 += v.x * b.x + v.y * b.y + v.z * b.z + v.w * b.w;
    }
    l0 += gb[0]; l1 += gb[1];
    float mx = fmaxf(l0, l1);
    float e0 = __expf(l0 - mx), e1 = __expf(l1 - mx);
    float inv = 0.5f / (e0 + e1);   // lambda = 0.5 folded in
    gateS[(size_t)i * 2 + 0] = e0 * inv;
    gateS[(size_t)i * 2 + 1] = e1 * inv;
}

// ---------------- pass A: attention logit per (edge, head); atomic max per dst
// wave per edge; lanes 0-15 head 0, lanes 16-31 head 1 (float4 slice each)
__global__ __launch_bounds__(256)
void edge_attn_max(const float* __restrict__ xl, const float* __restrict__ xr,
                   const int* __restrict__ src, const int* __restrict__ dst,
                   int nE, int nLoop, const float* __restrict__ att,
                   float* __restrict__ ebuf, float* __restrict__ mmax)
{
    int wave = (int)((blockIdx.x * blockDim.x + threadIdx.x) >> 5);
    int lane = threadIdx.x & 31;
    int total = nE + nLoop;
    if (wave >= total) return;
    int s, d;
    if (wave < nE) { s = src[wave]; d = dst[wave]; } else { s = wave - nE; d = s; }

    const int h   = lane >> 4;
    const int off = h * CH + (lane & 15) * 4;
    float4 a = *(const float4*)(xl + (size_t)s * HC + off);
    float4 b = *(const float4*)(xr + (size_t)d * HC + off);
    float4 w = *(const float4*)(att + off);
    float t, acc = 0.f;
    t = a.x + b.x; t = t > 0.f ? t : NEGSL * t; acc += t * w.x;
    t = a.y + b.y; t = t > 0.f ? t : NEGSL * t; acc += t * w.y;
    t = a.z + b.z; t = t > 0.f ? t : NEGSL * t; acc += t * w.z;
    t = a.w + b.w; t = t > 0.f ? t : NEGSL * t; acc += t * w.w;
    acc += __shfl_xor(acc, 1);
    acc += __shfl_xor(acc, 2);
    acc += __shfl_xor(acc, 4);
    acc += __shfl_xor(acc, 8);
    if ((lane & 15) == 0) {
        ebuf[(size_t)wave * 2 + h] = acc;
        __hip_atomic_fetch_max(&mmax[(size_t)d * 2 + h], acc,
                               __ATOMIC_RELAXED, __HIP_MEMORY_SCOPE_AGENT);
    }
}

// ---------------- pass B: ex = exp(e - m[dst]); denom += ex (both heads per thread)
__global__ __launch_bounds__(256)
void edge_expsum(const int* __restrict__ dst, int nE, int nLoop,
                 float* __restrict__ ebuf, const float* __restrict__ m,
                 float* __restrict__ den)
{
    int i = blockIdx.x * blockDim.x + threadIdx.x;
    int total = nE + nLoop;
    if (i >= total) return;
    int d = (i < nE) ? dst[i] : (i - nE);
    float2 e  = ((const float2*)ebuf)[i];
    float2 mm = ((const float2*)m)[d];
    float ex0 = __expf(e.x - mm.x);
    float ex1 = __expf(e.y - mm.y);
    ((float2*)ebuf)[i] = make_float2(ex0, ex1);
    atomicAdd(&den[(size_t)d * 2 + 0], ex0);
    atomicAdd(&den[(size_t)d * 2 + 1], ex1);
}

// ---------------- pass C: out[dst] += scale * 0.5 * (a0*xl[src,0,:] + a1*xl[src,1,:])
// wave per edge; each lane handles 2 output channels
__global__ __launch_bounds__(256)
void edge_msg(const float* __restrict__ xl,
              const int* __restrict__ src, const int* __restrict__ dst,
              int nE, int nLoop, const float* __restrict__ ebuf,
              const float* __restrict__ den, const float* __restrict__ gateS,
              int mode, float* __restrict__ out)
{
    int wave = (int)((blockIdx.x * blockDim.x + threadIdx.x) >> 5);
    int lane = threadIdx.x & 31;
    int total = nE + nLoop;
    if (wave >= total) return;
    int s, d;
    if (wave < nE) { s = src[wave]; d = dst[wave]; } else { s = wave - nE; d = s; }

    float2 ex = ((const float2*)ebuf)[wave];
    float2 dn = ((const float2*)den)[d];
    float a0 = ex.x / (dn.x + 1e-16f);
    float a1 = ex.y / (dn.y + 1e-16f);
    float hw = (mode == 0) ? 0.5f : gateS[(size_t)d * 2 + (mode - 1)]; // gateS has lambda
    float sc = 0.5f * hw;   // 0.5 = mean over heads

    float2 v0 = *(const float2*)(xl + (size_t)s * HC + lane * 2);        // head 0
    float2 v1 = *(const float2*)(xl + (size_t)s * HC + CH + lane * 2);   // head 1
    atomicAdd(&out[(size_t)d * CH + lane * 2 + 0], sc * (a0 * v0.x + a1 * v1.x));
    atomicAdd(&out[(size_t)d * CH + lane * 2 + 1], sc * (a0 * v0.y + a1 * v1.y));
}

// ---------------- init / epilogue ----------------
__global__ void zero_f32(float* p, int n)
{
    int i = blockIdx.x * blockDim.x + threadIdx.x;
    if (i < n) p[i] = 0.f;
}

__global__ void init_md(float* m, float* d, int n)
{
    int i = blockIdx.x * blockDim.x + threadIdx.x;
    if (i < n) { m[i] = -__builtin_inff(); d[i] = 0.f; }
}

__global__ void add_bias(float* __restrict__ out,
                         const float* __restrict__ b0, const float* __restrict__ b1,
                         const float* __restrict__ b2, const float* __restrict__ gateS,
                         int n)
{
    int i = blockIdx.x * blockDim.x + threadIdx.x;
    if (i >= n * CH) return;
    int node = i >> 6, c = i & 63;
    out[i] += 0.5f * b0[c]
            + gateS[(size_t)node * 2 + 0] * b1[c]
            + gateS[(size_t)node * 2 + 1] * b2[c];
}

// ---------------- launcher ----------------
extern "C" void kernel_launch(void* const* d_in, const int* in_sizes, int n_in,
                              void* d_out, int out_size, void* d_ws, size_t ws_size,
                              hipStream_t stream)
{
    const float* x     = (const float*)d_in[0];
    const float* wl    = (const float*)d_in[1];
    const float* bl    = (const float*)d_in[2];
    const float* wr    = (const float*)d_in[3];
    const float* br    = (const float*)d_in[4];
    const float* att[3]= {(const float*)d_in[5], (const float*)d_in[6], (const float*)d_in[7]};
    const float* bias0 = (const float*)d_in[8];
    const float* bias1 = (const float*)d_in[9];
    const float* bias2 = (const float*)d_in[10];
    const float* gw    = (const float*)d_in[11];
    const float* gb    = (const float*)d_in[12];
    const int*   ei[3] = {(const int*)d_in[13], (const int*)d_in[14], (const int*)d_in[15]};
    float* out = (float*)d_out;

    // workspace layout (floats): ~59 MB total
    float* wsf   = (float*)d_ws;
    float* xl    = wsf;                                     // N*128
    float* xr    = xl + (size_t)NNODES * HC;                // N*128
    float* gateS = xr + (size_t)NNODES * HC;                // N*2
    float* ebuf  = gateS + (size_t)NNODES * 2;              // (E+N)*2
    float* mbuf  = ebuf + (size_t)(NEDGES + NNODES) * 2;    // N*2
    float* dbuf  = mbuf + (size_t)NNODES * 2;               // N*2

    zero_f32<<<(NNODES * CH + 255) / 256, 256, 0, stream>>>(out, NNODES * CH);
    proj_wmma<<<NNODES / 16, 256, 0, stream>>>(x, wl, bl, wr, br, xl, xr);
    gate_kernel<<<(NNODES + 255) / 256, 256, 0, stream>>>(x, gw, gb, gateS, NNODES);

    for (int hop = 0; hop < 3; ++hop) {
        const int nLoop = (hop == 0) ? NNODES : 0;
        const int* src = ei[hop];
        const int* dst = ei[hop] + NEDGES;
        const int total = NEDGES + nLoop;
        init_md<<<(NNODES * 2 + 255) / 256, 256, 0, stream>>>(mbuf, dbuf, NNODES * 2);
        edge_attn_max<<<(total + 7) / 8, 256, 0, stream>>>(
            xl, xr, src, dst, NEDGES, nLoop, att[hop], ebuf, mbuf);
        edge_expsum<<<(total + 255) / 256, 256, 0, stream>>>(
            dst, NEDGES, nLoop, ebuf, mbuf, dbuf);
        edge_msg<<<(total + 7) / 8, 256, 0, stream>>>(
            xl, src, dst, NEDGES, nLoop, ebuf, dbuf, gateS, hop, out);
    }
    add_bias<<<(NNODES * CH + 255) / 256, 256, 0, stream>>>(
        out, bias0, bias1, bias2, gateS, NNODES);
}